// DiffReactParamPDE_85478439125433
// MI455X (gfx1250) — compile-verified
//
#include <hip/hip_runtime.h>
#include <cstdint>
#include <cmath>

// ---------------------------------------------------------------------------
// DiffReactParamPDE step, MI455X (gfx1250).
//
// Memory-bound stencil: 268 MB HBM traffic, ~0.4 GFLOP -> HBM floor ~11.5 us
// at 23.3 TB/s. WMMA cannot help (AI ~1.5 flop/byte). We use the CDNA5
// async global->LDS DMA path (GLOBAL_LOAD_ASYNC_TO_LDS_B128 / ASYNCcnt) to
// stage stencil rows into LDS, process R=4 grid rows per block (staging
// amplification 1.5x instead of 3x to relieve L2), and stream the output
// with non-temporal b128 stores so it doesn't evict the input set from L2.
// ---------------------------------------------------------------------------

#define NXD 4096
#define NYD 4096
#define RROWS 4               // output rows per block
#define BROWS (RROWS + 2)     // staged rows (with halo)

// 1/(DX*DX) with DX=0.1
#define INV_DX2 100.0f

typedef float v4f __attribute__((ext_vector_type(4)));

// Async DMA: copy 16 bytes from global 'g' into LDS byte-offset 'lds_byte'
// for this lane. Tracked by ASYNCcnt (CDNA5 ISA).
__device__ __forceinline__ void async_ld_b128(uint32_t lds_byte, const float* g) {
  asm volatile("global_load_async_to_lds_b128 %0, %1, off"
               :: "v"(lds_byte), "v"(g)
               : "memory");
}

__device__ __forceinline__ void wait_async_all() {
  asm volatile("s_wait_asynccnt 0" ::: "memory");
}

__device__ __forceinline__ float sigm01(float v) {
  return 0.01f * (1.0f / (1.0f + __expf(-v)));
}

__global__ __launch_bounds__(256)
void DiffReactPDE_step_kernel(const float* __restrict__ state,
                              const float* __restrict__ bc,
                              const float* __restrict__ a_org,
                              const float* __restrict__ b_org,
                              const float* __restrict__ k_org,
                              float* __restrict__ out)
{
  // LDS tile: [channel][6 rows: x0-1 .. x0+4][1024 columns] = 48 KB
  __shared__ float sm[2][BROWS][1024];

  const int t  = threadIdx.x;             // 0..255
  const int x0 = blockIdx.y * RROWS;      // first output row of this block
  const int yb = blockIdx.x << 10;        // 1024-column slab
  const int y0 = yb + (t << 2);           // this thread's 4 columns
  const size_t NXY = (size_t)NXD * (size_t)NYD;
  const int c4 = t << 2;

  const float* U = state;
  const float* V = state + NXY;

  // ---- Stage BROWS rows x 2 channels into LDS via async DMA ---------------
  // Staged row i holds global row clamp(x0-1+i); clamped rows' data is
  // replaced by bc values at compute time, so garbage there is harmless.
#pragma unroll
  for (int ch = 0; ch < 2; ++ch) {
    const float* base = state + (size_t)ch * NXY;
#pragma unroll
    for (int i = 0; i < BROWS; ++i) {
      int gx = x0 - 1 + i;
      gx = (gx < 0) ? 0 : ((gx > NXD - 1) ? (NXD - 1) : gx);
      const float* src = base + (size_t)gx * NYD + (size_t)y0;
      const uint32_t dst = (uint32_t)(uintptr_t)&sm[ch][i][c4];
      async_ld_b128(dst, src);
    }
  }
  wait_async_all();     // my wave's async ops complete
  __syncthreads();      // all waves' staging visible

  // ---- Uniform scalars ----------------------------------------------------
  const float a = sigm01(a_org[0]);
  const float b = sigm01(b_org[0]);
  const float k = sigm01(k_org[0]);

  // bc layout (1,2,4): [ch][left,right,top,bottom]; left/right pad y-axis,
  // top/bottom pad x-axis (reference _pad_bc).
  const float bcUL = bc[0], bcUR = bc[1], bcUT = bc[2], bcUB = bc[3];
  const float bcVL = bc[4], bcVR = bc[5], bcVT = bc[6], bcVB = bc[7];

  // ---- Compute RROWS rows -------------------------------------------------
#pragma unroll
  for (int r = 0; r < RROWS; ++r) {
    const int x = x0 + r;        // global row
    const int cb = r + 1;        // center row in staged buffer

    const v4f cU = *(const v4f*)&sm[0][cb][c4];
    const v4f cV = *(const v4f*)&sm[1][cb][c4];

    v4f nU, sU, nV, sV;
    if (x == 0) {
      nU = (v4f){bcUT, bcUT, bcUT, bcUT};
      nV = (v4f){bcVT, bcVT, bcVT, bcVT};
    } else {
      nU = *(const v4f*)&sm[0][cb - 1][c4];
      nV = *(const v4f*)&sm[1][cb - 1][c4];
    }
    if (x == NXD - 1) {
      sU = (v4f){bcUB, bcUB, bcUB, bcUB};
      sV = (v4f){bcVB, bcVB, bcVB, bcVB};
    } else {
      sU = *(const v4f*)&sm[0][cb + 1][c4];
      sV = *(const v4f*)&sm[1][cb + 1][c4];
    }

    // West/east single elements: LDS inside the slab, global/bc at edges.
    float wU, eU, wV, eV;
    if (t > 0)        { wU = sm[0][cb][c4 - 1]; wV = sm[1][cb][c4 - 1]; }
    else if (y0 == 0) { wU = bcUL;              wV = bcVL; }
    else              { wU = U[(size_t)x * NYD + (y0 - 1)];
                        wV = V[(size_t)x * NYD + (y0 - 1)]; }

    if (t < 255)            { eU = sm[0][cb][c4 + 4]; eV = sm[1][cb][c4 + 4]; }
    else if (y0 + 4 >= NYD) { eU = bcUR;              eV = bcVR; }
    else                    { eU = U[(size_t)x * NYD + (y0 + 4)];
                              eV = V[(size_t)x * NYD + (y0 + 4)]; }

    // 5-point Laplacians
    v4f lU, lV;
    lU.x = (nU.x + sU.x + wU   + cU.y - 4.0f * cU.x) * INV_DX2;
    lU.y = (nU.y + sU.y + cU.x + cU.z - 4.0f * cU.y) * INV_DX2;
    lU.z = (nU.z + sU.z + cU.y + cU.w - 4.0f * cU.z) * INV_DX2;
    lU.w = (nU.w + sU.w + cU.z + eU   - 4.0f * cU.w) * INV_DX2;

    lV.x = (nV.x + sV.x + wV   + cV.y - 4.0f * cV.x) * INV_DX2;
    lV.y = (nV.y + sV.y + cV.x + cV.z - 4.0f * cV.y) * INV_DX2;
    lV.z = (nV.z + sV.z + cV.y + cV.w - 4.0f * cV.z) * INV_DX2;
    lV.w = (nV.w + sV.w + cV.z + eV   - 4.0f * cV.w) * INV_DX2;

    // Reaction terms
    v4f du, dv;
    du.x = a * lU.x + cU.x - cU.x * cU.x * cU.x - cV.x - k;
    du.y = a * lU.y + cU.y - cU.y * cU.y * cU.y - cV.y - k;
    du.z = a * lU.z + cU.z - cU.z * cU.z * cU.z - cV.z - k;
    du.w = a * lU.w + cU.w - cU.w * cU.w * cU.w - cV.w - k;

    dv.x = b * lV.x + cU.x - cV.x;
    dv.y = b * lV.y + cU.y - cV.y;
    dv.z = b * lV.z + cU.z - cV.z;
    dv.w = b * lV.w + cU.w - cV.w;

    // Streaming output: non-temporal b128 stores (TH_STORE_NT) keep the
    // zero-reuse output from displacing the input set in L2.
    __builtin_nontemporal_store(du, (v4f*)&out[(size_t)x * NYD + (size_t)y0]);
    __builtin_nontemporal_store(dv, (v4f*)&out[NXY + (size_t)x * NYD + (size_t)y0]);
  }
}

extern "C" void kernel_launch(void* const* d_in, const int* in_sizes, int n_in,
                              void* d_out, int out_size, void* d_ws, size_t ws_size,
                              hipStream_t stream) {
  (void)in_sizes; (void)n_in; (void)out_size; (void)d_ws; (void)ws_size;
  const float* state = (const float*)d_in[0];
  const float* bc    = (const float*)d_in[1];
  const float* aorg  = (const float*)d_in[2];
  const float* borg  = (const float*)d_in[3];
  const float* korg  = (const float*)d_in[4];
  float* out = (float*)d_out;

  dim3 grid(NYD / 1024, NXD / RROWS, 1);  // 4 x 1024 blocks
  dim3 block(256, 1, 1);                  // 8 wave32 per block
  DiffReactPDE_step_kernel<<<grid, block, 0, stream>>>(state, bc, aorg, borg, korg, out);
}